// RefBasedDeepMetric_39651138077344
// MI455X (gfx1250) — compile-verified
//
#include <hip/hip_runtime.h>

typedef __attribute__((ext_vector_type(2))) float v2f;
typedef __attribute__((ext_vector_type(8))) float v8f;

#define B_ROWS 4096
#define R_ROWS 2048
#define K_DIM  2048
#define F_DIM  1024

// tiles: wave tile = 32x32. refs: 64x32 wave-tiles. x: 128x32 wave-tiles.
#define R_ROWTILES 64
#define X_ROWTILES 128
#define COLTILES   32

// ws layout (floats) -- every slot is fully written every call (no zeroing,
// no atomics, deterministic reduction order).
#define WS_CSPART 0                               // [64][1024] col-sum partials
#define WS_SSPART (WS_CSPART + R_ROWTILES * F_DIM)        // [2048] sumsq partials
#define WS_SQPART (WS_SSPART + R_ROWTILES * COLTILES)     // [32][4096] row |v|^2 partials
#define WS_DTPART (WS_SQPART + COLTILES * B_ROWS)         // [32][4096] row dot partials
#define WS_COLSUM (WS_DTPART + COLTILES * B_ROWS)         // [1024]
#define WS_SUMSQ  (WS_COLSUM + F_DIM)                     // [1]

// One wave computes a 32x32 tile of (A @ W) as a 2x2 grid of 16x16 WMMA tiles.
// fp32 WMMA, K stepped by 4 (unrolled x4). A-fragment: per-lane addressed v2f
// load (khalf folded into the address). B-fragment: two b32 column-strided
// loads, each fragment reused by two WMMAs.
struct Acc4 { v8f c00, c01, c10, c11; };

__device__ __forceinline__ Acc4 tile_gemm32x32_f32(const float* __restrict__ A,
                                                   const float* __restrict__ W,
                                                   int tM, int tN, int lane) {
    const int khalf = (lane >> 4) << 1;             // 0 (lanes 0-15) or 2 (16-31)
    const int l15 = lane & 15;
    const float* arow0 = A + (size_t)(tM + l15) * K_DIM + khalf;
    const float* arow1 = arow0 + (size_t)16 * K_DIM;
    const float* wcol0 = W + (size_t)khalf * F_DIM + (tN + l15);
    const float* wcol1 = wcol0 + 16;

    Acc4 r;
    r.c00 = (v8f){}; r.c01 = (v8f){}; r.c10 = (v8f){}; r.c11 = (v8f){};

    for (int k = 0; k < K_DIM; k += 16) {
#pragma unroll
        for (int kk = 0; kk < 16; kk += 4) {
            const int ko = k + kk;
            v2f a0 = *(const v2f*)(arow0 + ko);
            v2f a1 = *(const v2f*)(arow1 + ko);
            const float* b0p = wcol0 + (size_t)ko * F_DIM;
            const float* b1p = wcol1 + (size_t)ko * F_DIM;
            v2f b0, b1;
            b0.x = b0p[0]; b0.y = b0p[F_DIM];
            b1.x = b1p[0]; b1.y = b1p[F_DIM];
            r.c00 = __builtin_amdgcn_wmma_f32_16x16x4_f32(false, a0, false, b0, (short)0, r.c00, false, false);
            r.c01 = __builtin_amdgcn_wmma_f32_16x16x4_f32(false, a0, false, b1, (short)0, r.c01, false, false);
            r.c10 = __builtin_amdgcn_wmma_f32_16x16x4_f32(false, a1, false, b0, (short)0, r.c10, false, false);
            r.c11 = __builtin_amdgcn_wmma_f32_16x16x4_f32(false, a1, false, b1, (short)0, r.c11, false, false);
        }
    }
    return r;
}

// vrefs = refs @ W + b ; store per-tile column-sum and sumsq partials.
__global__ __launch_bounds__(256) void vrefs_stats_kernel(
    const float* __restrict__ refs, const float* __restrict__ W,
    const float* __restrict__ bias, float* __restrict__ cs_part,
    float* __restrict__ ss_part) {
    const int lane = threadIdx.x & 31;
    const int wave = threadIdx.x >> 5;
    const int tile = blockIdx.x * 8 + wave;          // 64 row-tiles x 32 col-tiles
    const int rt = tile >> 5;                        // row supertile 0..63
    const int tM = rt << 5;
    const int tN = (tile & 31) << 5;
    const int col0 = tN + (lane & 15);
    const int col1 = col0 + 16;

    Acc4 r = tile_gemm32x32_f32(refs, W, tM, tN, lane);

    const float bv0 = bias[col0];
    const float bv1 = bias[col1];
    float cs0 = 0.0f, cs1 = 0.0f, ss = 0.0f;
#pragma unroll
    for (int v = 0; v < 8; ++v) {
        float e00 = r.c00[v] + bv0, e01 = r.c01[v] + bv1;
        float e10 = r.c10[v] + bv0, e11 = r.c11[v] + bv1;
        cs0 += e00 + e10;
        cs1 += e01 + e11;
        ss  += e00 * e00 + e01 * e01 + e10 * e10 + e11 * e11;
    }
    // fold the two half-waves (same column in both halves)
    cs0 += __shfl_xor(cs0, 16, 32);
    cs1 += __shfl_xor(cs1, 16, 32);
    if (lane < 16) {                                  // unique (rt, col) owner
        cs_part[rt * F_DIM + col0] = cs0;
        cs_part[rt * F_DIM + col1] = cs1;
    }
#pragma unroll
    for (int m = 16; m >= 1; m >>= 1) ss += __shfl_xor(ss, m, 32);
    if (lane == 0) ss_part[tile] = ss;                // unique per tile
}

// Deterministic reduction of refs-side partials: 4 blocks do columns,
// block 4 does the sumsq scalar.
__global__ __launch_bounds__(256) void refs_reduce_kernel(
    const float* __restrict__ cs_part, const float* __restrict__ ss_part,
    float* __restrict__ col_sum, float* __restrict__ sum_sq) {
    if (blockIdx.x < 4) {
        int c = blockIdx.x * 256 + threadIdx.x;       // 0..1023
        float s = 0.0f;
        for (int t = 0; t < R_ROWTILES; ++t) s += cs_part[t * F_DIM + c];
        col_sum[c] = s;
    } else {
        __shared__ float red[256];
        float s = 0.0f;
        for (int i = threadIdx.x; i < R_ROWTILES * COLTILES; i += 256)
            s += ss_part[i];
        red[threadIdx.x] = s;
        __syncthreads();
        for (int w = 128; w >= 1; w >>= 1) {
            if (threadIdx.x < w) red[threadIdx.x] += red[threadIdx.x + w];
            __syncthreads();
        }
        if (threadIdx.x == 0) sum_sq[0] = red[0];
    }
}

// vin = x @ W + b ; store per-(coltile,row) partials of |vin|^2 and vin.vmean.
__global__ __launch_bounds__(256) void vin_accum_kernel(
    const float* __restrict__ x, const float* __restrict__ W,
    const float* __restrict__ bias, const float* __restrict__ col_sum,
    float* __restrict__ sq_part, float* __restrict__ dt_part) {
    const int lane = threadIdx.x & 31;
    const int wave = threadIdx.x >> 5;
    const int tile = blockIdx.x * 8 + wave;          // 128 row-tiles x 32 col-tiles
    const int ct = tile & 31;                        // col supertile 0..31
    const int tM = (tile >> 5) << 5;
    const int tN = ct << 5;
    const int col0 = tN + (lane & 15);
    const int col1 = col0 + 16;

    Acc4 r = tile_gemm32x32_f32(x, W, tM, tN, lane);

    const float bv0 = bias[col0];
    const float bv1 = bias[col1];
    const float vm0 = col_sum[col0] * (1.0f / (float)R_ROWS);
    const float vm1 = col_sum[col1] * (1.0f / (float)R_ROWS);
    const int mbase = tM + ((lane >> 4) << 3);       // lanes 16-31 hold rows M=v+8
    float* sqp = sq_part + (size_t)ct * B_ROWS;
    float* dtp = dt_part + (size_t)ct * B_ROWS;

#pragma unroll
    for (int v = 0; v < 8; ++v) {
        // rows tM .. tM+15 come from c00/c01
        float e0 = r.c00[v] + bv0, e1 = r.c01[v] + bv1;
        float sqA = e0 * e0 + e1 * e1;
        float dtA = e0 * vm0 + e1 * vm1;
        // rows tM+16 .. tM+31 come from c10/c11
        float f0 = r.c10[v] + bv0, f1 = r.c11[v] + bv1;
        float sqB = f0 * f0 + f1 * f1;
        float dtB = f0 * vm0 + f1 * vm1;
#pragma unroll
        for (int m = 8; m >= 1; m >>= 1) {            // reduce across 16 columns
            sqA += __shfl_xor(sqA, m, 32);
            dtA += __shfl_xor(dtA, m, 32);
            sqB += __shfl_xor(sqB, m, 32);
            dtB += __shfl_xor(dtB, m, 32);
        }
        if ((lane & 15) == 0) {                       // unique (ct, row) owner
            sqp[mbase + v]      = sqA;
            dtp[mbase + v]      = dtA;
            sqp[mbase + v + 16] = sqB;
            dtp[mbase + v + 16] = dtB;
        }
    }
}

// Sum the 32 column-supertile partials per row (fixed order) and finalize.
__global__ void finalize_kernel(const float* __restrict__ sq_part,
                                const float* __restrict__ dt_part,
                                const float* __restrict__ sum_sq,
                                float* __restrict__ out) {
    int b = blockIdx.x * blockDim.x + threadIdx.x;
    if (b >= B_ROWS) return;
    float sq = 0.0f, dt = 0.0f;
    for (int t = 0; t < COLTILES; ++t) {
        sq += sq_part[(size_t)t * B_ROWS + b];
        dt += dt_part[(size_t)t * B_ROWS + b];
    }
    out[b] = sq + sum_sq[0] * (1.0f / (float)R_ROWS) - 2.0f * dt;
}

extern "C" void kernel_launch(void* const* d_in, const int* in_sizes, int n_in,
                              void* d_out, int out_size, void* d_ws, size_t ws_size,
                              hipStream_t stream) {
    const float* x    = (const float*)d_in[0];   // [4096, 2048]
    const float* refs = (const float*)d_in[1];   // [2048, 2048]
    const float* W    = (const float*)d_in[2];   // [2048, 1024]
    const float* bias = (const float*)d_in[3];   // [1024]
    float* out = (float*)d_out;                  // [4096]

    float* ws      = (float*)d_ws;
    float* cs_part = ws + WS_CSPART;
    float* ss_part = ws + WS_SSPART;
    float* sq_part = ws + WS_SQPART;
    float* dt_part = ws + WS_DTPART;
    float* col_sum = ws + WS_COLSUM;
    float* sum_sq  = ws + WS_SUMSQ;

    // refs: 64*32 = 2048 wave-tiles / 8 waves = 256 blocks
    vrefs_stats_kernel<<<256, 256, 0, stream>>>(refs, W, bias, cs_part, ss_part);
    refs_reduce_kernel<<<5, 256, 0, stream>>>(cs_part, ss_part, col_sum, sum_sq);
    // x: 128*32 = 4096 wave-tiles / 8 waves = 512 blocks
    vin_accum_kernel<<<512, 256, 0, stream>>>(x, W, bias, col_sum, sq_part, dt_part);
    finalize_kernel<<<(B_ROWS + 255) / 256, 256, 0, stream>>>(sq_part, dt_part, sum_sq, out);
}